// RandomForestClassifier_84439057039520
// MI455X (gfx1250) — compile-verified
//
#include <hip/hip_runtime.h>
#include <hip/hip_bf16.h>

// out[r] = x[r,:] . mean_e(W[e,0,:]) + mean_e(b[e])   (fp32 end-to-end)
// Memory-bound: 160 MB read + 16 MB write @ 23.3 TB/s -> ~7.6 us floor.
// Dot product runs on the matrix pipe via V_WMMA_F32_16X16X4_F32:
//   A (16x4)  = broadcast mean-weight rows (loop-invariant, built once)
//   B (4x16)  = 16-row tile of x, K-major (three b64 loads per lane)
//   D (16x16) = every row m holds the 16 tile outputs -> coalesced 64B store.

typedef __attribute__((ext_vector_type(2))) float v2f;
typedef __attribute__((ext_vector_type(8))) float v8f;

#define D_IN 10
#define N_EST 10

// ws[0..15] = zero-padded mean weight vector, ws[16] = mean bias.
// The zero padding (k=10..15) is what makes the main kernel's k=10,11
// B-operand garbage harmless (multiplied by A=0).
__global__ void rf_setup_kernel(const float* __restrict__ W,
                                const float* __restrict__ b,
                                float* __restrict__ ws) {
    int t = threadIdx.x;
    if (t < 16) {
        float s = 0.0f;
        if (t < D_IN) {
            #pragma unroll
            for (int e = 0; e < N_EST; ++e) s += W[e * D_IN + t];
        }
        ws[t] = s * (1.0f / N_EST);
    } else if (t == 16) {
        float s = 0.0f;
        #pragma unroll
        for (int e = 0; e < N_EST; ++e) s += b[e];
        ws[16] = s * (1.0f / N_EST);
    }
}

__global__ __launch_bounds__(256) void rf_main_kernel(const float* __restrict__ x,
                                                      const float* __restrict__ ws,
                                                      float* __restrict__ out,
                                                      int nrows, int ntiles,
                                                      int total_waves) {
    const int lane   = threadIdx.x & 31;
    const int waveId = (int)((blockIdx.x * blockDim.x + threadIdx.x) >> 5);
    const int half   = lane >> 4;     // 0: lanes 0-15 (K even pair), 1: lanes 16-31 (K odd pair)
    const int r      = lane & 15;     // row within tile / output column n

    // ---- loop-invariant setup (once per wave) ----
    float wv[12];
    #pragma unroll
    for (int k = 0; k < 12; ++k) wv[k] = ws[k];   // wv[10..11] == 0
    const float bavg = ws[16];

    // A-matrix 16x4 f32 layout: lane<16 holds {K=4s+0 (v0), K=4s+1 (v1)},
    // lane>=16 holds {K=4s+2, K=4s+3}. A[m][k] = wv[k] for every m.
    v2f a0, a1, a2;
    a0.x = wv[0 + 2 * half];  a0.y = wv[1 + 2 * half];
    a1.x = wv[4 + 2 * half];  a1.y = wv[5 + 2 * half];
    a2.x = wv[8 + 2 * half];  a2.y = wv[9 + 2 * half];   // zero for half==1 -> B garbage ok

    // ---- grid-stride over 16-row tiles ----
    for (int t = waveId; t < ntiles; t += total_waves) {
        const long base = (long)t * 16;
        long row = base + r;
        if (row >= (long)nrows) row = (long)nrows - 1;   // safety (4e6 % 16 == 0)
        const float* xr = x + row * D_IN;

        // B-matrix 4x16 f32 layout mirrors A's K distribution per lane-half.
        // B[k][n] = x[base+n][k]; lane n (and n+16) read row base+n.
        v2f b0, b1, b2;
        b0.x = xr[0 + 2 * half];  b0.y = xr[1 + 2 * half];   // K 0..3
        b1.x = xr[4 + 2 * half];  b1.y = xr[5 + 2 * half];   // K 4..7
        b2.x = xr[8];             b2.y = xr[9];              // K 8..9 (10,11 x A==0)

        v8f c = {};
        c = __builtin_amdgcn_wmma_f32_16x16x4_f32(false, a0, false, b0, (short)0, c, false, false);
        c = __builtin_amdgcn_wmma_f32_16x16x4_f32(false, a1, false, b1, (short)0, c, false, false);
        c = __builtin_amdgcn_wmma_f32_16x16x4_f32(false, a2, false, b2, (short)0, c, false, false);

        // D[0][n] lives in VGPR 0, lane n (n = 0..15): coalesced 64B store.
        if (r == lane && base + r < (long)nrows) {       // lanes 0-15 only
            out[base + r] = c[0] + bavg;
        }
    }
}

extern "C" void kernel_launch(void* const* d_in, const int* in_sizes, int n_in,
                              void* d_out, int out_size, void* d_ws, size_t ws_size,
                              hipStream_t stream) {
    const float* x = (const float*)d_in[0];   // [B, 10]
    const float* W = (const float*)d_in[1];   // [10, 1, 10]
    const float* b = (const float*)d_in[2];   // [10, 1]
    float* out = (float*)d_out;               // [B, 1]
    float* ws  = (float*)d_ws;                // >= 17 floats

    const int nrows  = in_sizes[0] / D_IN;
    const int ntiles = (nrows + 15) / 16;     // 250,000 for B = 4M

    rf_setup_kernel<<<1, 32, 0, stream>>>(W, b, ws);

    const int waves_per_block = 8;            // 256 threads, wave32
    int blocks = (ntiles + waves_per_block - 1) / waves_per_block;
    if (blocks > 2048) blocks = 2048;         // grid-stride: ~15 tiles per wave
    const int total_waves = blocks * waves_per_block;

    rf_main_kernel<<<blocks, 256, 0, stream>>>(x, ws, out, nrows, ntiles, total_waves);
}